// Criterion_67954972557706
// MI455X (gfx1250) — compile-verified
//
#include <hip/hip_runtime.h>
#include <hip/hip_bf16.h>
#include <math.h>

// ---------------------------------------------------------------------------
// Problem constants (from reference)
// ---------------------------------------------------------------------------
#define BB       2
#define N_CLOTH  16384
#define M_OBS    12000
#define N_FACES  16384
#define EPS_C    1e-3f
#define WSTART   1e-3f
#define WMAX     5e3f

typedef float v2f __attribute__((ext_vector_type(2)));
typedef float v8f __attribute__((ext_vector_type(8)));
typedef int   v4i __attribute__((ext_vector_type(4)));

typedef __attribute__((address_space(1))) v4i g_v4i;   // global int4
typedef __attribute__((address_space(3))) v4i l_v4i;   // LDS int4

// Async global->LDS 16-byte copy (per-lane), ASYNCcnt-tracked.
// Signature (from hipcc diagnostic): (v4i AS1*, v4i AS3*, imm offset, imm cpol)
__device__ __forceinline__ void async_copy_b128(const float4* g, float4* l) {
#if __has_builtin(__builtin_amdgcn_global_load_async_to_lds_b128)
    __builtin_amdgcn_global_load_async_to_lds_b128(
        (g_v4i*)g, (l_v4i*)l, 0, 0);
#else
    unsigned lds_addr = (unsigned)(__UINTPTR_TYPE__)(l_v4i*)l;
    unsigned long long gaddr = (unsigned long long)(__UINTPTR_TYPE__)g;
    asm volatile("global_load_async_to_lds_b128 %0, %1, off"
                 :: "v"(lds_addr), "v"(gaddr) : "memory");
#endif
}

__device__ __forceinline__ void wait_asynccnt_le4() {
    asm volatile("s_wait_asynccnt 0x4" ::: "memory");
}
__device__ __forceinline__ void wait_asynccnt_0() {
    asm volatile("s_wait_asynccnt 0x0" ::: "memory");
}

// ---------------------------------------------------------------------------
// Kernel 1: per-face precompute.
//   bcurr[f]  = (-2*cc.x, -2*cc.y, -2*cc.z, |cc|^2)  from CURRENT obstacle pos
//   nrmd0[f]  = (n.x, n.y, n.z, dot(cn, n))          from NEXT obstacle pos
// ---------------------------------------------------------------------------
__global__ __launch_bounds__(256) void prep_faces_kernel(
    const float* __restrict__ obs_curr,   // [B, M, 3]
    const float* __restrict__ obs_next,   // [B, M, 3]
    const int*   __restrict__ faces,      // [B, F, 3]
    float4* __restrict__ bcurr,           // [B*F]
    float4* __restrict__ nrmd0)           // [B*F]
{
    int g = blockIdx.x * blockDim.x + threadIdx.x;
    if (g >= BB * N_FACES) return;
    int b = g / N_FACES;

    const int* fv = faces + (size_t)g * 3;
    int i0 = fv[0], i1 = fv[1], i2 = fv[2];
    i0 = (i0 < 0) ? 0 : (i0 >= M_OBS ? M_OBS - 1 : i0);
    i1 = (i1 < 0) ? 0 : (i1 >= M_OBS ? M_OBS - 1 : i1);
    i2 = (i2 < 0) ? 0 : (i2 >= M_OBS ? M_OBS - 1 : i2);

    // --- current positions: centroid -> distance-score vector ---
    {
        const float* vc = obs_curr + (size_t)b * M_OBS * 3;
        float cx = (vc[i0*3+0] + vc[i1*3+0] + vc[i2*3+0]) * (1.0f/3.0f);
        float cy = (vc[i0*3+1] + vc[i1*3+1] + vc[i2*3+1]) * (1.0f/3.0f);
        float cz = (vc[i0*3+2] + vc[i1*3+2] + vc[i2*3+2]) * (1.0f/3.0f);
        bcurr[g] = make_float4(-2.0f*cx, -2.0f*cy, -2.0f*cz,
                               cx*cx + cy*cy + cz*cz);
    }

    // --- next positions: centroid + unit normal -> plane (n, d0) ---
    {
        const float* vn = obs_next + (size_t)b * M_OBS * 3;
        float ax = vn[i0*3+0], ay = vn[i0*3+1], az = vn[i0*3+2];
        float bx = vn[i1*3+0], by = vn[i1*3+1], bz = vn[i1*3+2];
        float gx = vn[i2*3+0], gy = vn[i2*3+1], gz = vn[i2*3+2];
        float cx = (ax + bx + gx) * (1.0f/3.0f);
        float cy = (ay + by + gy) * (1.0f/3.0f);
        float cz = (az + bz + gz) * (1.0f/3.0f);
        float e1x = bx - ax, e1y = by - ay, e1z = bz - az;
        float e2x = gx - ax, e2y = gy - ay, e2z = gz - az;
        float nx = e1y*e2z - e1z*e2y;
        float ny = e1z*e2x - e1x*e2z;
        float nz = e1x*e2y - e1y*e2x;
        float inv = 1.0f / (sqrtf(nx*nx + ny*ny + nz*nz) + 1e-12f);
        nx *= inv; ny *= inv; nz *= inv;
        float d0 = cx*nx + cy*ny + cz*nz;
        nrmd0[g] = make_float4(nx, ny, nz, d0);
    }
}

// ---------------------------------------------------------------------------
// Kernel 2: brute-force 1-NN via V_WMMA_F32_16X16X4_F32 with double-buffered
// async global->LDS staging (GLOBAL_LOAD_ASYNC_TO_LDS_B128 + s_wait_asynccnt).
// Score(n,f) = -2*p_n·c_f + |c_f|^2  ==  A(16x4) x B(4x16) with
//   A row  = [x, y, z, 1],  B col = [-2cx, -2cy, -2cz, |c|^2]
// A layout (ISA 7.12.2): lanes 0-15 hold (K0,K1), lanes 16-31 hold (K2,K3),
// row M = lane&15.  B mirrors A.  D: VGPR j -> row j (low lanes) / j+8 (high),
// col = lane&15.  One wave owns 16 cloth rows; 4 waves share LDS face chunks.
// ---------------------------------------------------------------------------
#define CHUNK_FACES 512   // 8 KB per buffer, x2 buffers = 16 KB LDS

__global__ __launch_bounds__(128) void nn_wmma_kernel(
    const float*  __restrict__ cloth_curr, // [B, N, 3]
    const float4* __restrict__ bcurr,      // [B*F]
    int*          __restrict__ nn_idx)     // [B*N]
{
    const int blocksPerBatch = N_CLOTH / 64;           // 64 rows per block
    const int b       = blockIdx.x / blocksPerBatch;
    const int tileBlk = blockIdx.x % blocksPerBatch;
    const int tid  = threadIdx.x;
    const int wave = tid >> 5;
    const int lane = tid & 31;
    const int lo   = lane & 15;
    const bool hiHalf = (lane >= 16);
    const int m0  = tileBlk * 64 + wave * 16;
    const int row = m0 + lo;

    // Build the A operand (persistent across the whole loop).
    const float* p = cloth_curr + ((size_t)b * N_CLOTH + row) * 3;
    float px = p[0], py = p[1], pz = p[2];
    v2f a;
    a.x = hiHalf ? pz   : px;   // K2 : K0
    a.y = hiHalf ? 1.0f : py;   // K3 : K1

    __shared__ float4 sf[2][CHUNK_FACES];
    const float4* bc = bcurr + (size_t)b * N_FACES;

    float minv[8];
    int   mini[8];
#pragma unroll
    for (int j = 0; j < 8; ++j) { minv[j] = 3.4e38f; mini[j] = 0; }

    const int nch = N_FACES / CHUNK_FACES;

    // Prologue: kick off chunk 0 into buffer 0 (4 x b128 per thread).
#pragma unroll
    for (int k = 0; k < 4; ++k)
        async_copy_b128(&bc[tid + k * 128], &sf[0][tid + k * 128]);

    for (int c = 0; c < nch; ++c) {
        const int pbuf  = c & 1;
        const int fbase = c * CHUNK_FACES;

        if (c + 1 < nch) {
            // Issue next chunk's DMA before computing on the current one.
            const int nbase = fbase + CHUNK_FACES;
#pragma unroll
            for (int k = 0; k < 4; ++k)
                async_copy_b128(&bc[nbase + tid + k * 128],
                                &sf[1 - pbuf][tid + k * 128]);
            // Async loads complete in order: <=4 outstanding means the 4
            // transfers of the *current* chunk have landed in LDS.
            wait_asynccnt_le4();
        } else {
            wait_asynccnt_0();
        }
        __syncthreads();

#pragma unroll 4
        for (int t = 0; t < CHUNK_FACES / 16; ++t) {
            float4 fb = sf[pbuf][t * 16 + lo];
            v2f bt;
            bt.x = hiHalf ? fb.z : fb.x;  // K2 : K0
            bt.y = hiHalf ? fb.w : fb.y;  // K3 : K1

            v8f cc = {};
            v8f d = __builtin_amdgcn_wmma_f32_16x16x4_f32(
                false, a, false, bt, (short)0, cc, false, false);

            int fidx = fbase + t * 16 + lo;
#pragma unroll
            for (int j = 0; j < 8; ++j) {
                float v = d[j];
                if (v < minv[j]) { minv[j] = v; mini[j] = fidx; }
            }
        }
        __syncthreads();   // reads done before this buffer is re-filled
    }

    // Cross-lane argmin within each 16-lane half (columns of the D tile).
#pragma unroll
    for (int j = 0; j < 8; ++j) {
#pragma unroll
        for (int m = 8; m >= 1; m >>= 1) {
            float ov = __shfl_xor(minv[j], m, 32);
            int   oi = __shfl_xor(mini[j], m, 32);
            if (ov < minv[j] || (ov == minv[j] && oi < mini[j])) {
                minv[j] = ov; mini[j] = oi;
            }
        }
    }

    if (lo == 0) {
        int rbase = m0 + (hiHalf ? 8 : 0);
#pragma unroll
        for (int j = 0; j < 8; ++j)
            nn_idx[(size_t)b * N_CLOTH + rbase + j] = mini[j];
    }
}

// ---------------------------------------------------------------------------
// Kernel 3a: zero the scalar loss slot (d_out is poisoned by the harness).
// ---------------------------------------------------------------------------
__global__ void zero_loss_kernel(float* __restrict__ out) {
    if (threadIdx.x == 0 && blockIdx.x == 0) out[0] = 0.0f;
}

// ---------------------------------------------------------------------------
// Kernel 3b: signed distance -> cubic penalty -> per-vertex + scalar loss.
// out[0] = loss, out[1..N] = per_vert.
// ---------------------------------------------------------------------------
__global__ __launch_bounds__(256) void finalize_kernel(
    const float*  __restrict__ cloth_next, // [B, N, 3]
    const float4* __restrict__ nrmd0,      // [B*F]
    const int*    __restrict__ nn_idx,     // [B*N]
    const int*    __restrict__ iter_num,   // [1]
    float*        __restrict__ out)        // [1 + N]
{
    int n = blockIdx.x * blockDim.x + threadIdx.x;

    float it = (float)iter_num[0] - 50000.0f;
    if (it < 0.0f) it = 0.0f;
    float progress = it * 1e-5f;
    if (progress > 1.0f) progress = 1.0f;
    float weight = WSTART + (WMAX - WSTART) * progress;

    float pv = 0.0f;
    if (n < N_CLOTH) {
#pragma unroll
        for (int b = 0; b < BB; ++b) {
            int idx = nn_idx[(size_t)b * N_CLOTH + n];
            float4 nd = nrmd0[(size_t)b * N_FACES + idx];
            const float* pn = cloth_next + ((size_t)b * N_CLOTH + n) * 3;
            float dist = pn[0]*nd.x + pn[1]*nd.y + pn[2]*nd.z - nd.w;
            float t = EPS_C - dist;
            if (t < 0.0f) t = 0.0f;
            pv += t * t * t;
        }
        pv *= weight * (1.0f / BB);
        out[1 + n] = pv;
    }

    __shared__ float red[256];
    red[threadIdx.x] = pv;
    __syncthreads();
    for (int s = 128; s > 0; s >>= 1) {
        if (threadIdx.x < s) red[threadIdx.x] += red[threadIdx.x + s];
        __syncthreads();
    }
    if (threadIdx.x == 0) atomicAdd(out, red[0]);
}

// ---------------------------------------------------------------------------
// Host entry
// ---------------------------------------------------------------------------
extern "C" void kernel_launch(void* const* d_in, const int* in_sizes, int n_in,
                              void* d_out, int out_size, void* d_ws, size_t ws_size,
                              hipStream_t stream) {
    const float* cloth_curr = (const float*)d_in[0];
    const float* cloth_next = (const float*)d_in[1];
    const float* obs_curr   = (const float*)d_in[2];
    const float* obs_next   = (const float*)d_in[3];
    const int*   faces      = (const int*)d_in[4];
    const int*   iter_num   = (const int*)d_in[5];
    float* out = (float*)d_out;

    // Workspace layout (16B aligned):
    //   bcurr  : B*F float4   (512 KB)
    //   nrmd0  : B*F float4   (512 KB)
    //   nn_idx : B*N int      (128 KB)
    char* ws = (char*)d_ws;
    float4* bcurr = (float4*)ws;
    float4* nrmd0 = (float4*)(ws + (size_t)BB * N_FACES * sizeof(float4));
    int* nn_idx   = (int*)  (ws + (size_t)2 * BB * N_FACES * sizeof(float4));

    // 1) per-face precompute
    {
        int total = BB * N_FACES;
        prep_faces_kernel<<<(total + 255) / 256, 256, 0, stream>>>(
            obs_curr, obs_next, faces, bcurr, nrmd0);
    }

    // 2) WMMA nearest-face search (64 cloth rows per block, 4 waves)
    {
        int blocks = BB * (N_CLOTH / 64);
        nn_wmma_kernel<<<blocks, 128, 0, stream>>>(cloth_curr, bcurr, nn_idx);
    }

    // 3) finalize
    zero_loss_kernel<<<1, 32, 0, stream>>>(out);
    finalize_kernel<<<(N_CLOTH + 255) / 256, 256, 0, stream>>>(
        cloth_next, nrmd0, nn_idx, iter_num, out);
}